// Scene_68719476736203
// MI455X (gfx1250) — compile-verified
//
#include <hip/hip_runtime.h>

// ---------------------------------------------------------------------------
// Constants derived from the reference
// ---------------------------------------------------------------------------
#define IMRES   128
#define MESHW   20
#define NVERT   400
#define NTRI    722          // 2 * 19 * 19
#define TPAD    736          // 46 chunks of 16
#define NCHUNK  46
#define BIGF    1.0e9f
#define LDSF2   (3 * 2 * TPAD)        // float2 entries staged per block (4416)
#define LDSWORDS (2 * LDSF2)          // 8832 dwords = 35328 bytes

__device__ __constant__ const float FOCAL = 11.430052302761343f;   // 1/tan(5 deg)
__device__ __constant__ const float DISTC = 5.7150261513806715f;   // 0.5/tan(5 deg)

typedef __attribute__((ext_vector_type(2))) float v2f;
typedef __attribute__((ext_vector_type(8))) float v8f;
typedef unsigned int u32x4 __attribute__((ext_vector_type(4)));
typedef int          i32x4 __attribute__((ext_vector_type(4)));
typedef int          i32x8 __attribute__((ext_vector_type(8)));

#if defined(__gfx1250__) && __has_builtin(__builtin_amdgcn_wmma_f32_16x16x4_f32)
#define USE_WMMA 1
#endif
#if defined(__gfx1250__) && __has_builtin(__builtin_amdgcn_tensor_load_to_lds) && \
    __has_builtin(__builtin_amdgcn_s_wait_tensorcnt)
#define USE_TDM 1
#endif

// frag layout: float2 entries, index ((b*3 + q)*2 + half)*TPAD + tri
//   q: 0=w0 coeffs, 1=w1 coeffs, 2=depth coeffs
//   half 0 -> (Ax, Ay)  (A-matrix K=0,1 for lanes 0-15)
//   half 1 -> (Ac, 0 )  (A-matrix K=2,3 for lanes 16-31)

// ---------------------------------------------------------------------------
// Kernel 1: blur xyz (on the fly) + bilinear sample at the 400 mesh vertices
// ---------------------------------------------------------------------------
__global__ void prep_verts(const float* __restrict__ xyz, float* __restrict__ verts) {
    int id = blockIdx.x * blockDim.x + threadIdx.x;
    if (id >= NVERT * 3) return;
    int v  = id / 3;
    int ch = id % 3;
    int vi = v / MESHW, vj = v % MESHW;
    float u  = (float)vj * (1.0f / 19.0f);
    float vv = (float)vi * (1.0f / 19.0f);
    float x = u  * 127.0f;
    float y = vv * 127.0f;
    float x0f = fminf(fmaxf(floorf(x), 0.0f), 126.0f);
    float y0f = fminf(fmaxf(floorf(y), 0.0f), 126.0f);
    int x0 = (int)x0f, y0 = (int)y0f;
    float fx = x - x0f, fy = y - y0f;

    // normalized gaussian, sigma=2, radius=6
    float k[13];
    float ksum = 0.0f;
    #pragma unroll
    for (int i = 0; i < 13; ++i) {
        float d = (float)(i - 6) * 0.5f;
        k[i] = __expf(-0.5f * d * d);
        ksum += k[i];
    }
    float inv2 = 1.0f / (ksum * ksum);

    float t[4];  // (y0,x0) (y0,x0+1) (y0+1,x0) (y0+1,x0+1)
    #pragma unroll
    for (int q = 0; q < 4; ++q) {
        int ty = y0 + (q >> 1);
        int tx = x0 + (q & 1);
        float acc = 0.0f;
        for (int a = 0; a < 13; ++a) {
            int sy = min(max(ty + a - 6, 0), 127);
            float row = 0.0f;
            for (int c = 0; c < 13; ++c) {
                int sx = min(max(tx + c - 6, 0), 127);
                row += k[c] * xyz[(sy * 128 + sx) * 3 + ch];
            }
            acc += k[a] * row;
        }
        t[q] = acc * inv2;
    }
    verts[id] = (t[0] * (1.0f - fx) + t[1] * fx) * (1.0f - fy)
              + (t[2] * (1.0f - fx) + t[3] * fx) * fy;
}

// ---------------------------------------------------------------------------
// Kernel 2: per (batch, triangle) affine coefficients, in WMMA A-fragment order
// ---------------------------------------------------------------------------
__global__ void prep_tris(const float* __restrict__ angles,
                          const float* __restrict__ verts,
                          float2* __restrict__ frag) {
    int id = blockIdx.x * blockDim.x + threadIdx.x;
    if (id >= 2 * TPAD) return;
    int b = id / TPAD;
    int t = id % TPAD;

    auto put = [&](int q, float ax, float ay, float ac) {
        frag[((b * 3 + q) * 2 + 0) * TPAD + t] = make_float2(ax, ay);
        frag[((b * 3 + q) * 2 + 1) * TPAD + t] = make_float2(ac, 0.0f);
    };

    if (t >= NTRI) { put(0, 0, 0, -1.0f); put(1, 0, 0, -1.0f); put(2, 0, 0, 0.0f); return; }

    // triangle -> vertex ids (sphere grid connectivity)
    int va, vb, vc;
    if (t < 361) { int i = t / 19, j = t % 19;
        va = i * MESHW + j; vb = (i + 1) * MESHW + j; vc = i * MESHW + j + 1; }
    else { int t2 = t - 361; int i = t2 / 19, j = t2 % 19;
        va = (i + 1) * MESHW + j; vb = (i + 1) * MESHW + j + 1; vc = i * MESHW + j + 1; }

    // rotation: angle ranges [-60,60]x[-90,90]x[0,0] deg -> R = Ry * Rx (Rz = I)
    float a0 = angles[b * 3 + 0] * 2.0943951023931953f - 1.0471975511965976f;
    float a1 = angles[b * 3 + 1] * 3.1415926535897931f - 1.5707963267948966f;
    float cx = cosf(a0), sx = sinf(a0), cy = cosf(a1), sy = sinf(a1);

    float nx[3], ny[3], nz[3];
    int vid[3] = {va, vb, vc};
    #pragma unroll
    for (int q = 0; q < 3; ++q) {
        float X = verts[vid[q] * 3 + 0];
        float Y = verts[vid[q] * 3 + 1];
        float Z = verts[vid[q] * 3 + 2];
        float Xc =  cy * X + sy * sx * Y + sy * cx * Z;
        float Yc =           cx * Y      - sx * Z;
        float Zc = -sy * X + cy * sx * Y + cy * cx * Z + DISTC;
        nx[q] = FOCAL * Xc / Zc;
        ny[q] = FOCAL * Yc / Zc;
        nz[q] = Zc;
    }
    float area = (nx[1] - nx[0]) * (ny[2] - ny[0]) - (ny[1] - ny[0]) * (nx[2] - nx[0]);
    if (!(fabsf(area) > 1e-6f)) { put(0, 0, 0, -1.0f); put(1, 0, 0, -1.0f); put(2, 0, 0, 0.0f); return; }

    // w0 = edge(v1,v2)/area : affine A*px + B*py + C
    float e0x = nx[2] - nx[1], e0y = ny[2] - ny[1];
    float A0 = -e0y / area, B0 = e0x / area, C0 = (e0y * nx[1] - e0x * ny[1]) / area;
    // w1 = edge(v2,v0)/area
    float e1x = nx[0] - nx[2], e1y = ny[0] - ny[2];
    float A1 = -e1y / area, B1 = e1x / area, C1 = (e1y * nx[2] - e1x * ny[2]) / area;
    // depth = z2 + w0*(z0-z2) + w1*(z1-z2)
    float dz0 = nz[0] - nz[2], dz1 = nz[1] - nz[2];
    float Ad = A0 * dz0 + A1 * dz1;
    float Bd = B0 * dz0 + B1 * dz1;
    float Cd = C0 * dz0 + C1 * dz1 + nz[2];

    put(0, A0, B0, C0);
    put(1, A1, B1, C1);
    put(2, Ad, Bd, Cd);
}

// ---------------------------------------------------------------------------
// Bilinear texture fetch (HWC float32, clamped like the reference)
// ---------------------------------------------------------------------------
__device__ inline float3 bilin3(const float* __restrict__ tex, int H, int W,
                                float u, float v) {
    float x = u * (float)(W - 1);
    float y = v * (float)(H - 1);
    float x0f = fminf(fmaxf(floorf(x), 0.0f), (float)(W - 2));
    float y0f = fminf(fmaxf(floorf(y), 0.0f), (float)(H - 2));
    int x0 = (int)x0f, y0 = (int)y0f;
    float fx = x - x0f, fy = y - y0f;
    const float* p00 = tex + (y0 * W + x0) * 3;
    const float* p01 = tex + (y0 * W + x0 + 1) * 3;
    const float* p10 = tex + ((y0 + 1) * W + x0) * 3;
    const float* p11 = tex + ((y0 + 1) * W + x0 + 1) * 3;
    float3 r;
    r.x = (p00[0] * (1 - fx) + p01[0] * fx) * (1 - fy) + (p10[0] * (1 - fx) + p11[0] * fx) * fy;
    r.y = (p00[1] * (1 - fx) + p01[1] * fx) * (1 - fy) + (p10[1] * (1 - fx) + p11[1] * fx) * fy;
    r.z = (p00[2] * (1 - fx) + p01[2] * fx) * (1 - fy) + (p10[2] * (1 - fx) + p11[2] * fx) * fy;
    return r;
}

// ---------------------------------------------------------------------------
// Kernel 3: TDM -> LDS -> WMMA rasterizer.
// The per-batch coefficient table (35328 B) is staged once per workgroup into
// LDS by the Tensor Data Mover, then V_WMMA_F32_16X16X4_F32 evaluates
// barycentrics + depth for 16 tris x 16 pixels per instruction from ds loads.
// ---------------------------------------------------------------------------
__global__ void __launch_bounds__(256)
raster(const float2* __restrict__ frag, const float* __restrict__ angles,
       const float* __restrict__ rgb, const float* __restrict__ bcg,
       float* __restrict__ out) {
    extern __shared__ float2 lds[];   // LDSF2 float2 = 35328 bytes, LDS offset 0

    int lane = threadIdx.x & 31;
    int wave = threadIdx.x >> 5;
    int tile = blockIdx.x * (blockDim.x >> 5) + wave;   // 0..2047
    int b    = tile >> 10;                              // 1024 tiles / batch
    int tpix = (tile & 1023) << 4;
    int col  = lane & 15;
    int half = lane >> 4;
    int pix  = tpix + col;
    int iy = pix >> 7, jx = pix & 127;
    float px = ((float)jx + 0.5f) * (1.0f / 64.0f) - 1.0f;
    float py = ((float)iy + 0.5f) * (1.0f / 64.0f) - 1.0f;

    // ---- stage this batch's coefficient table into LDS -------------------
    const float2* gsrc = frag + (size_t)b * LDSF2;
#if defined(USE_TDM)
    if (wave == 0) {
        unsigned long long ga = (unsigned long long)(uintptr_t)gsrc;
        // D# group 0: count=1, lds_addr=0 (dynamic LDS base), global addr, type=2
        u32x4 g0;
        g0.x = 1u;                                            // count = 1 (valid)
        g0.y = 0u;                                            // lds_addr = 0
        g0.z = (unsigned)(ga & 0xFFFFFFFFu);                  // global_addr[31:0]
        g0.w = (unsigned)((ga >> 32) & 0x01FFFFFFu) | (2u << 30); // addr[56:32] | type=2
        // D# group 1: 1-D tile of LDSWORDS 4-byte elements
        i32x8 g1;
        g1[0] = 0x00020000;                    // wg_mask=0, data_size=2 (4 bytes)
        g1[1] = (int)((unsigned)LDSWORDS << 16);   // tensor_dim0[15:0] in [31:16]
        g1[2] = (int)(1u << 16);               // tensor_dim0 hi=0, tensor_dim1=1
        g1[3] = (int)((unsigned)LDSWORDS << 16);   // tile_dim0 = LDSWORDS
        g1[4] = 0;                             // tile_dim1 = tile_dim2 = 0 (1-D)
        g1[5] = LDSWORDS;                      // tensor_dim0_stride[31:0]
        g1[6] = 0;
        g1[7] = 0;
        i32x4 gz = {0, 0, 0, 0};
#if __clang_major__ >= 23
        i32x8 gz8 = {0, 0, 0, 0, 0, 0, 0, 0};
        __builtin_amdgcn_tensor_load_to_lds(g0, g1, gz, gz, gz8, 0);
#else
        __builtin_amdgcn_tensor_load_to_lds(g0, g1, gz, gz, 0);
#endif
        __builtin_amdgcn_s_wait_tensorcnt(0);
    }
    __syncthreads();
#else
    for (int i = threadIdx.x; i < LDSF2; i += blockDim.x) lds[i] = gsrc[i];
    __syncthreads();
#endif

#if defined(USE_WMMA)
    // B fragment (4x16, K-major): lanes 0-15 hold K0/K1 = (px,py),
    // lanes 16-31 hold K2/K3 = (1,0).
    v2f Bm;
    Bm.x = half ? 1.0f : px;
    Bm.y = half ? 0.0f : py;
#endif

    float bestD = BIGF;
    int   bestT = 0;

    int base0 = (0 + half) * TPAD;   // q=0
    int base1 = (2 + half) * TPAD;   // q=1
    int base2 = (4 + half) * TPAD;   // q=2

    for (int c = 0; c < NCHUNK; ++c) {
#if defined(USE_WMMA)
        int tri = c * 16 + col;                          // A row M = col
        float2 f0 = lds[base0 + tri];
        float2 f1 = lds[base1 + tri];
        float2 f2 = lds[base2 + tri];
        v2f A0 = {f0.x, f0.y};
        v2f A1 = {f1.x, f1.y};
        v2f A2 = {f2.x, f2.y};
        v8f z = {0.f, 0.f, 0.f, 0.f, 0.f, 0.f, 0.f, 0.f};
        v8f W0 = __builtin_amdgcn_wmma_f32_16x16x4_f32(false, A0, false, Bm, (short)0, z, false, false);
        v8f W1 = __builtin_amdgcn_wmma_f32_16x16x4_f32(false, A1, false, Bm, (short)0, z, false, false);
        v8f DP = __builtin_amdgcn_wmma_f32_16x16x4_f32(false, A2, false, Bm, (short)0, z, false, false);
        int rbase = c * 16 + half * 8;                   // D row M = r + 8*half
        #pragma unroll
        for (int r = 0; r < 8; ++r) {
            float w0 = W0[r], w1 = W1[r];
            float w2 = 1.0f - w0 - w1;
            // inside <=> no sign bit set on any barycentric (single or3 + cmp)
            int sg = __float_as_int(w0) | __float_as_int(w1) | __float_as_int(w2);
            float d = (sg >= 0) ? DP[r] : BIGF;
            if (d < bestD) { bestD = d; bestT = rbase + r; }
        }
#else
        // scalar fallback: every lane evaluates all 16 tris of the chunk
        for (int r = 0; r < 16; ++r) {
            int tri2 = c * 16 + r;
            float2 a0 = lds[0 * TPAD + tri2];
            float2 c0 = lds[1 * TPAD + tri2];
            float2 a1 = lds[2 * TPAD + tri2];
            float2 c1 = lds[3 * TPAD + tri2];
            float2 a2 = lds[4 * TPAD + tri2];
            float2 c2 = lds[5 * TPAD + tri2];
            float w0 = a0.x * px + a0.y * py + c0.x;
            float w1 = a1.x * px + a1.y * py + c1.x;
            float w2 = 1.0f - w0 - w1;
            int sg = __float_as_int(w0) | __float_as_int(w1) | __float_as_int(w2);
            float d = (sg >= 0) ? (a2.x * px + a2.y * py + c2.x) : BIGF;
            if (d < bestD) { bestD = d; bestT = tri2; }
        }
#endif
    }

    // merge the two half-wave partial argmins (same pixel column, lane ^ 16)
    float oD = __shfl_xor(bestD, 16, 32);
    int   oT = __shfl_xor(bestT, 16, 32);
    if (oD < bestD || (oD == bestD && oT < bestT)) { bestD = oD; bestT = oT; }

    if (lane >= 16) return;   // lanes 0-15 own the 16 pixels of this tile

    // recompute the winning barycentrics from the LDS coefficients
    float2 a0xy = lds[0 * TPAD + bestT];
    float2 a0c  = lds[1 * TPAD + bestT];
    float2 a1xy = lds[2 * TPAD + bestT];
    float2 a1c  = lds[3 * TPAD + bestT];
    float bw0 = a0xy.x * px + a0xy.y * py + a0c.x;
    float bw1 = a1xy.x * px + a1xy.y * py + a1c.x;
    float w2b = 1.0f - bw0 - bw1;
    bool  mask = bestD < BIGF;

    // UVs of the winning triangle, regenerated from the mesh formula
    int t = bestT, va, vb, vc;
    if (t < 361) { int i = t / 19, j = t % 19;
        va = i * MESHW + j; vb = (i + 1) * MESHW + j; vc = i * MESHW + j + 1; }
    else { int t2 = t - 361; int i = t2 / 19, j = t2 % 19;
        va = (i + 1) * MESHW + j; vb = (i + 1) * MESHW + j + 1; vc = i * MESHW + j + 1; }
    const float s = 1.0f / 19.0f;
    float u0 = (va % MESHW) * s, v0 = (va / MESHW) * s;
    float u1 = (vb % MESHW) * s, v1 = (vb / MESHW) * s;
    float u2 = (vc % MESHW) * s, v2 = (vc / MESHW) * s;
    float uu = bw0 * u0 + bw1 * u1 + w2b * u2;
    float vv = bw0 * v0 + bw1 * v1 + w2b * v2;
    uu = fminf(fmaxf(uu, 0.0f), 1.0f);
    vv = fminf(fmaxf(vv, 0.0f), 1.0f);
    float3 color = bilin3(rgb, 128, 128, uu, vv);

    float minb  = fminf(fminf(bw0, bw1), w2b);
    float alpha = mask ? 1.0f / (1.0f + __expf(-minb * 20.0f)) : 0.0f;  // MARGIN=0.05

    // background: rh = rw = 128/384
    const float rwh = 128.0f / 384.0f;
    float y1c = (1.0f - rwh) * angles[b * 3 + 0];
    float x1c = (1.0f - rwh) * angles[b * 3 + 1];
    float bu = fminf(fmaxf(x1c + rwh * ((float)jx * (1.0f / 127.0f)), 0.0f), 1.0f);
    float bv = fminf(fmaxf(y1c + rwh * ((float)iy * (1.0f / 127.0f)), 0.0f), 1.0f);
    float3 bg = bilin3(bcg, 384, 384, bu, bv);

    int o = (b * (IMRES * IMRES) + pix) * 3;
    out[o + 0] = alpha * color.x + (1.0f - alpha) * bg.x;
    out[o + 1] = alpha * color.y + (1.0f - alpha) * bg.y;
    out[o + 2] = alpha * color.z + (1.0f - alpha) * bg.z;
}

// ---------------------------------------------------------------------------
// Launch
// ---------------------------------------------------------------------------
extern "C" void kernel_launch(void* const* d_in, const int* in_sizes, int n_in,
                              void* d_out, int out_size, void* d_ws, size_t ws_size,
                              hipStream_t stream) {
    const float* angles = (const float*)d_in[0];   // (2,3)
    const float* xyz    = (const float*)d_in[1];   // (1,128,128,3)
    const float* rgb    = (const float*)d_in[2];   // (1,128,128,3)
    const float* bcg    = (const float*)d_in[3];   // (1,384,384,3)
    float* out = (float*)d_out;                    // (2,128,128,3)

    float*  verts = (float*)d_ws;                                  // 1200 floats
    float2* frag  = (float2*)((char*)d_ws + 1200 * sizeof(float)); // 2*3*2*736 float2

    prep_verts<<<dim3(5),   dim3(256), 0, stream>>>(xyz, verts);
    prep_tris <<<dim3(6),   dim3(256), 0, stream>>>(angles, verts, frag);
    raster    <<<dim3(256), dim3(256), LDSF2 * sizeof(float2), stream>>>(frag, angles, rgb, bcg, out);
}